// VNNBlock_48954037240412
// MI455X (gfx1250) — compile-verified
//
#include <hip/hip_runtime.h>
#include <math.h>

typedef float v2f __attribute__((ext_vector_type(2)));
typedef float v8f __attribute__((ext_vector_type(8)));

// freq_tab[k] = 10000^(-k/16)  (the sinusoidal-encoding frequency for pair k)
__constant__ const float FREQ_TAB[16] = {
    1.0f,            0.56234133f,     0.31622777f,     0.17782794f,
    0.1f,            0.056234133f,    0.031622777f,    0.017782794f,
    0.01f,           0.0056234133f,   0.0031622777f,   0.0017782794f,
    0.001f,          0.00056234133f,  0.00031622777f,  0.00017782794f};

// ---------------------------------------------------------------------------
// Kernel 1: xsum[s] = sum_i x[s,i]   (one wave per sample)
// ---------------------------------------------------------------------------
__global__ __launch_bounds__(32) void vnn_xsum(const float* __restrict__ x,
                                               float* __restrict__ xsum) {
    const int s = blockIdx.x;
    const int lane = threadIdx.x;
    float v = x[s * 128 + lane] + x[s * 128 + lane + 32] +
              x[s * 128 + lane + 64] + x[s * 128 + lane + 96];
    for (int off = 16; off; off >>= 1) v += __shfl_xor(v, off, 32);
    if (lane == 0) xsum[s] = v;
}

// ---------------------------------------------------------------------------
// Kernel 2: 128x32x128 f32 GEMM via V_WMMA_F32_16X16X4_F32.
//   BIASNN=false:  Q[o,h] = sum_d pe(o)[d]*(w1[d,h]+w1[d+32,h]) + b1[h]
//   BIASNN=true :  P[o,h] = sum_d pe(o)[d]*bw1[d,h]             + bb1[h]
// One wave per 16x16 output tile, straight-line 8-step K-loop (no branches).
// A fragment generated in-register: per step one angle -> (sin, cos) pair,
// frequency from a compile-time table (no exp in the loop).
// ---------------------------------------------------------------------------
template <bool BIASNN>
__global__ __launch_bounds__(256) void vnn_gemm_pe(const float* __restrict__ w,
                                                   const float* __restrict__ bias,
                                                   float* __restrict__ dst) {
    const int lane = threadIdx.x & 31;
    const int wid  = blockIdx.x * (blockDim.x >> 5) + (threadIdx.x >> 5);
    const int ti = wid >> 3, tj = wid & 7;   // 8x8 grid of 16x16 tiles
    const int half = lane >> 4;              // lanes 16-31 hold K-values +2
    const int l15  = lane & 15;
    const float row = (float)(ti * 16 + l15); // A row = position o
    const int col  = tj * 16 + l15;           // B/D col = hidden h

    v8f acc = {};
    #pragma unroll
    for (int k4 = 0; k4 < 8; ++k4) {
        // A frag (16x4 f32): VGPR0 = K{0|2} (even d -> sin), VGPR1 = K{1|3} (odd d -> cos)
        const float ang = row * FREQ_TAB[2 * k4 + half];
        v2f a, b;
        a.x = __sinf(ang);
        a.y = __cosf(ang);
        // B frag (4x16 f32): rows k0, k0+1 at this lane's column
        const int k0 = k4 * 4 + half * 2;
        if (BIASNN) {
            b.x = w[k0 * 128 + col];
            b.y = w[(k0 + 1) * 128 + col];
        } else {
            b.x = w[k0 * 128 + col]       + w[(k0 + 32) * 128 + col];
            b.y = w[(k0 + 1) * 128 + col] + w[(k0 + 33) * 128 + col];
        }
        // D = A*B + C   (neg_a, A, neg_b, B, c_mod, C, reuse_a, reuse_b)
        acc = __builtin_amdgcn_wmma_f32_16x16x4_f32(false, a, false, b,
                                                    (short)0, acc, false, false);
    }

    const float bv = bias[col];
    #pragma unroll
    for (int r = 0; r < 8; ++r) {            // C/D: VGPR r -> M = r (+8 for hi lanes)
        const int m = ti * 16 + r + half * 8;
        dst[m * 128 + col] = acc[r] + bv;
    }
}

// ---------------------------------------------------------------------------
// Kernel 3: one wave32 per (s,o) pair.  8192 pairs = 1024 blocks x 8 waves.
//   h  = relu(Q[o,:] + x[s,o]*w1[64,:]);  out = xsum[s]*(h.w2 + b2)
//   hb = relu(P[o,:] + out   *bw1[32,:]); res = out + hb.bw2 + bb2
// Vectorized: each lane owns 4 consecutive hidden units (one b128 load/array).
// ---------------------------------------------------------------------------
__global__ __launch_bounds__(256) void vnn_main(const float* __restrict__ x,
                                                const float* __restrict__ w1,
                                                const float* __restrict__ w2,
                                                const float* __restrict__ b2,
                                                const float* __restrict__ bw1,
                                                const float* __restrict__ bw2,
                                                const float* __restrict__ bb2,
                                                const float* __restrict__ Q,
                                                const float* __restrict__ P,
                                                const float* __restrict__ xsum,
                                                float* __restrict__ out) {
    const int lane = threadIdx.x & 31;
    const int gw = blockIdx.x * 8 + (threadIdx.x >> 5);
    const int s = gw >> 7;
    const int o = gw & 127;

    const float xso = x[s * 128 + o];
    const float4 q   = *(const float4*)(Q   + o * 128 + lane * 4);
    const float4 wl  = *(const float4*)(w1  + 64 * 128 + lane * 4);
    const float4 v2  = *(const float4*)(w2  + lane * 4);

    float accw = fmaf(fmaxf(fmaf(xso, wl.x, q.x), 0.f), v2.x,
                 fmaf(fmaxf(fmaf(xso, wl.y, q.y), 0.f), v2.y,
                 fmaf(fmaxf(fmaf(xso, wl.z, q.z), 0.f), v2.z,
                      fmaxf(fmaf(xso, wl.w, q.w), 0.f) * v2.w)));
    for (int off = 16; off; off >>= 1) accw += __shfl_xor(accw, off, 32);
    const float outv = xsum[s] * (accw + b2[0]);

    const float4 p   = *(const float4*)(P   + o * 128 + lane * 4);
    const float4 bl  = *(const float4*)(bw1 + 32 * 128 + lane * 4);
    const float4 u2  = *(const float4*)(bw2 + lane * 4);

    float accb = fmaf(fmaxf(fmaf(outv, bl.x, p.x), 0.f), u2.x,
                 fmaf(fmaxf(fmaf(outv, bl.y, p.y), 0.f), u2.y,
                 fmaf(fmaxf(fmaf(outv, bl.z, p.z), 0.f), u2.z,
                      fmaxf(fmaf(outv, bl.w, p.w), 0.f) * u2.w)));
    for (int off = 16; off; off >>= 1) accb += __shfl_xor(accb, off, 32);

    if (lane == 0) out[s * 128 + o] = outv + (accb + bb2[0]);
}

// ---------------------------------------------------------------------------
// Host launcher
// inputs: 0=x[64,128] 1=w1[65,128] 2=b1[128] 3=w2[128] 4=b2[1]
//         5=bw1[33,128] 6=bb1[128] 7=bw2[128] 8=bb2[1] 9=output_size 10=chunks
// output: [64,128] f32
// ---------------------------------------------------------------------------
extern "C" void kernel_launch(void* const* d_in, const int* in_sizes, int n_in,
                              void* d_out, int out_size, void* d_ws, size_t ws_size,
                              hipStream_t stream) {
    const float* x   = (const float*)d_in[0];
    const float* w1  = (const float*)d_in[1];
    const float* b1  = (const float*)d_in[2];
    const float* w2  = (const float*)d_in[3];
    const float* b2  = (const float*)d_in[4];
    const float* bw1 = (const float*)d_in[5];
    const float* bb1 = (const float*)d_in[6];
    const float* bw2 = (const float*)d_in[7];
    const float* bb2 = (const float*)d_in[8];
    float* out = (float*)d_out;

    float* ws = (float*)d_ws;
    float* Q    = ws;              // 128*128
    float* P    = ws + 16384;      // 128*128
    float* xsum = ws + 32768;      // 64

    vnn_xsum<<<64, 32, 0, stream>>>(x, xsum);
    vnn_gemm_pe<false><<<8, 256, 0, stream>>>(w1, b1, Q);   // weight_nn layer-1
    vnn_gemm_pe<true ><<<8, 256, 0, stream>>>(bw1, bb1, P); // bias_nn   layer-1
    vnn_main<<<1024, 256, 0, stream>>>(x, w1, w2, b2, bw1, bw2, bb2, Q, P, xsum, out);
}